// SparseMatrix_38414187496059
// MI455X (gfx1250) — compile-verified
//
#include <hip/hip_runtime.h>

#define HCOLS 64
#define BLOCK 256
#define TILE  256   // edges staged per LDS tile (== BLOCK)

// ---- CDNA5 async global->LDS path (ASYNCcnt), with safe fallbacks ----------
#if defined(__has_builtin)
#  if __has_builtin(__builtin_amdgcn_global_load_async_to_lds_b32)
#    define HAVE_ASYNC_LDS 1
#  endif
#  if __has_builtin(__builtin_amdgcn_s_wait_asynccnt)
#    define HAVE_WAIT_ASYNC 1
#  endif
#endif
#ifndef HAVE_ASYNC_LDS
#  define HAVE_ASYNC_LDS 0
#endif
#ifndef HAVE_WAIT_ASYNC
#  define HAVE_WAIT_ASYNC 0
#endif

typedef __attribute__((address_space(1))) int GI;  // global int
typedef __attribute__((address_space(3))) int LI;  // LDS int

__device__ __forceinline__ void async_copy_b32(const void* g, void* l) {
#if HAVE_ASYNC_LDS
  __builtin_amdgcn_global_load_async_to_lds_b32((GI*)g, (LI*)l, 0, 0);
#else
  *(volatile int*)l = *(const volatile int*)g;  // synchronous fallback
#endif
}

template <int N>
__device__ __forceinline__ void wait_async_le() {
#if HAVE_ASYNC_LDS
#  if HAVE_WAIT_ASYNC
  __builtin_amdgcn_s_wait_asynccnt(N);
#  else
  asm volatile("s_wait_asynccnt %0" ::"i"(N) : "memory");
#  endif
#else
  __threadfence_block();
#endif
}

// ---------------------------------------------------------------------------
__global__ void __launch_bounds__(BLOCK) spmm_zero_kernel(float* __restrict__ p,
                                                          int n) {
  int i = (blockIdx.x * BLOCK + threadIdx.x) * 4;
  if (i + 3 < n) {
    *reinterpret_cast<float4*>(p + i) = make_float4(0.f, 0.f, 0.f, 0.f);
  } else {
    for (; i < n; ++i) p[i] = 0.f;
  }
}

// COO SpMM: C[row[e], :] += edata[e] * B[col[e], :]
// One wave32 per edge; lane handles 2 columns (float2). Edge triples are
// double-buffered in LDS via async global->LDS copies tracked by ASYNCcnt.
__global__ void __launch_bounds__(BLOCK) spmm_coo_kernel(
    const float* __restrict__ edata, const int* __restrict__ row,
    const int* __restrict__ col, const float* __restrict__ B,
    float* __restrict__ C, int nnz, int ntiles) {
  __shared__ float s_val[2][TILE];
  __shared__ int s_row[2][TILE];
  __shared__ int s_col[2][TILE];

  const int t = threadIdx.x;
  const int lane = t & 31;
  const int wv = t >> 5;

  const int tile0 = blockIdx.x;
  if (tile0 >= ntiles) return;

  // Prefetch first tile (always exactly 3 async ops per wave: clamp indices).
  {
    int e = tile0 * TILE + t;
    if (e >= nnz) e = nnz - 1;
    async_copy_b32(&edata[e], &s_val[0][t]);
    async_copy_b32(&row[e], &s_row[0][t]);
    async_copy_b32(&col[e], &s_col[0][t]);
  }

  int buf = 0;
  for (int tile = tile0; tile < ntiles; tile += gridDim.x, buf ^= 1) {
    const int nbuf = buf ^ 1;
    const int ntile = tile + gridDim.x;
    const bool has_next = (ntile < ntiles);

    if (has_next) {
      int e = ntile * TILE + t;
      if (e >= nnz) e = nnz - 1;
      async_copy_b32(&edata[e], &s_val[nbuf][t]);
      async_copy_b32(&row[e], &s_row[nbuf][t]);
      async_copy_b32(&col[e], &s_col[nbuf][t]);
      wait_async_le<3>();  // current tile's 3 ops retired; next 3 in flight
    } else {
      wait_async_le<0>();
    }
    __syncthreads();  // tile data visible to all waves

    const int base = tile * TILE;
    int valid = nnz - base;
    if (valid > TILE) valid = TILE;

    const int jbeg = wv * 32;
    int jend = jbeg + 32;
    if (jend > valid) jend = valid;

    for (int j = jbeg; j < jend; ++j) {
      int r = s_row[buf][j];
      int c = s_col[buf][j];
      int wbits = __float_as_int(s_val[buf][j]);
      // These are wave-uniform (j is uniform) -> force into SGPRs so the
      // B/C addressing becomes saddr + lane offset.
      r = __builtin_amdgcn_readfirstlane(r);
      c = __builtin_amdgcn_readfirstlane(c);
      const float w = __int_as_float(__builtin_amdgcn_readfirstlane(wbits));

      const float2 b =
          *reinterpret_cast<const float2*>(B + (size_t)c * HCOLS + (lane << 1));
      float* cp = C + (size_t)r * HCOLS + (lane << 1);
      atomicAdd(cp + 0, w * b.x);
      atomicAdd(cp + 1, w * b.y);
    }
    __syncthreads();  // everyone done with buf before it is overwritten
  }
}

// ---------------------------------------------------------------------------
extern "C" void kernel_launch(void* const* d_in, const int* in_sizes, int n_in,
                              void* d_out, int out_size, void* d_ws,
                              size_t ws_size, hipStream_t stream) {
  (void)n_in;
  (void)d_ws;
  (void)ws_size;

  const float* edata = (const float*)d_in[0];
  const int* row = (const int*)d_in[1];
  const int* col = (const int*)d_in[2];
  const float* B = (const float*)d_in[3];
  float* C = (float*)d_out;

  const int nnz = in_sizes[0];

  // Zero the output (harness poisons it with 0xAA before timing).
  {
    const int nthreads = (out_size + 3) / 4;
    const int grid = (nthreads + BLOCK - 1) / BLOCK;
    spmm_zero_kernel<<<grid, BLOCK, 0, stream>>>(C, out_size);
  }

  const int ntiles = (nnz + TILE - 1) / TILE;
  int grid = ntiles < 2500 ? ntiles : 2500;
  if (grid < 1) grid = 1;
  spmm_coo_kernel<<<grid, BLOCK, 0, stream>>>(edata, row, col, B, C, nnz,
                                              ntiles);
}